// Sparsemax_81432579932948
// MI455X (gfx1250) — compile-verified
//
#include <hip/hip_runtime.h>
#include <hip/hip_bf16.h>

// Sparsemax along last dim (D2=1024) of [16,2048,1024] fp32.
// One wave32 per row; async global->LDS staging (CDNA5 ASYNCcnt path);
// candidate-compaction bisection instead of a full sort.

#define D2            1024
#define LANES         32
#define REGS          (D2 / LANES)        // 32 floats per lane
#define WAVES_PER_BLK 8
#define BLK_THREADS   (WAVES_PER_BLK * LANES)

__global__ __launch_bounds__(BLK_THREADS, 2)
void sparsemax1024_kernel(const float* __restrict__ x,
                          float* __restrict__ out,
                          int rows) {
    // 8 waves x 1024 floats = 32 KB LDS; buffer is reused for candidate compaction.
    __shared__ __align__(16) float srow[WAVES_PER_BLK][D2];

    const int lane = threadIdx.x & (LANES - 1);
    const int wave = threadIdx.x >> 5;
    const int row  = blockIdx.x * WAVES_PER_BLK + wave;
    if (row >= rows) return;   // wave-uniform exit

    const float* __restrict__ rowp = x + (size_t)row * D2;
    float* myrow = &srow[wave][0];

    // ---------------- Stage 1: async copy row global -> LDS ----------------
    // 8 x (32 lanes x 16B) = 4 KB. GV addressing mode (64-bit VGPR address).
    #pragma unroll
    for (int c = 0; c < 8; ++c) {
        unsigned int ldsAddr =
            (unsigned int)(size_t)(myrow + c * 128 + lane * 4);
        unsigned long long gaddr =
            (unsigned long long)(size_t)(rowp + c * 128 + lane * 4);
        asm volatile("global_load_async_to_lds_b128 %0, %1, off"
                     :: "v"(ldsAddr), "v"(gaddr)
                     : "memory");
    }
    asm volatile("s_wait_asynccnt 0" ::: "memory");

    // ---------------- Stage 2: LDS -> 32 registers per lane -----------------
    float y[REGS];
    #pragma unroll
    for (int c = 0; c < 8; ++c) {
        float4 v = *(const float4*)(myrow + c * 128 + lane * 4);  // ds_load_b128
        y[c * 4 + 0] = v.x; y[c * 4 + 1] = v.y;
        y[c * 4 + 2] = v.z; y[c * 4 + 3] = v.w;
    }

    // ---------------- Stage 3: row max, shift -------------------------------
    float m = y[0];
    #pragma unroll
    for (int j = 1; j < REGS; ++j) m = fmaxf(m, y[j]);
    #pragma unroll
    for (int off = 16; off >= 1; off >>= 1)
        m = fmaxf(m, __shfl_xor(m, off, LANES));
    #pragma unroll
    for (int j = 0; j < REGS; ++j) y[j] -= m;

    // ---------------- Stage 4: compact candidates (y > -1) into LDS ---------
    // tau* in [-1, 0], so support subset of {y > -1}. Deterministic prefix-sum
    // compaction (no atomics -> reproducible across graph replays).
    unsigned int cloc = 0;
    #pragma unroll
    for (int j = 0; j < REGS; ++j) cloc += (y[j] > -1.0f) ? 1u : 0u;

    unsigned int pre = cloc;                   // inclusive wave prefix sum
    #pragma unroll
    for (int off = 1; off < LANES; off <<= 1) {
        unsigned int t = __shfl_up(pre, off, LANES);
        if (lane >= off) pre += t;
    }
    const unsigned int base = pre - cloc;            // exclusive prefix
    const unsigned int C    = __shfl(pre, LANES - 1, LANES); // total candidates

    unsigned int o = base;
    #pragma unroll
    for (int j = 0; j < REGS; ++j)
        if (y[j] > -1.0f) myrow[o++] = y[j];          // reuse row buffer

    // Cache the (overwhelmingly common) first-32 candidates in a register.
    const float v0 = (lane < C) ? myrow[lane] : -2.0f;

    // ---------------- Stage 5: bisection on tau over candidates -------------
    float lo = -1.0f, hi = 0.0f;
    #pragma unroll 1
    for (int it = 0; it < 26; ++it) {
        const float tau = 0.5f * (lo + hi);
        float s = (v0 > tau) ? (v0 - tau) : 0.0f;
        for (unsigned int i = lane + LANES; i < C; i += LANES) {  // rare spill
            const float v = myrow[i];
            s += (v > tau) ? (v - tau) : 0.0f;
        }
        #pragma unroll
        for (int off = 16; off >= 1; off >>= 1)
            s += __shfl_xor(s, off, LANES);
        if (s >= 1.0f) lo = tau; else hi = tau;       // wave-uniform
    }

    // Exact refinement: tau = (sum_{y>tau_b} y - 1) / k
    {
        const float taub = 0.5f * (lo + hi);
        float S = 0.0f; unsigned int k = 0;
        if (v0 > taub) { S += v0; ++k; }
        for (unsigned int i = lane + LANES; i < C; i += LANES) {
            const float v = myrow[i];
            if (v > taub) { S += v; ++k; }
        }
        #pragma unroll
        for (int off = 16; off >= 1; off >>= 1) {
            S += __shfl_xor(S, off, LANES);
            k += __shfl_xor(k, off, LANES);
        }
        const float tau = (S - 1.0f) / (float)(k ? k : 1u);
        lo = tau;  // reuse lo as final tau
    }
    const float tau = lo;

    // ---------------- Stage 6: write out max(0, y - tau) --------------------
    float* __restrict__ orow = out + (size_t)row * D2;
    #pragma unroll
    for (int c = 0; c < 8; ++c) {
        float4 r;
        r.x = fmaxf(0.0f, y[c * 4 + 0] - tau);
        r.y = fmaxf(0.0f, y[c * 4 + 1] - tau);
        r.z = fmaxf(0.0f, y[c * 4 + 2] - tau);
        r.w = fmaxf(0.0f, y[c * 4 + 3] - tau);
        *(float4*)(orow + c * 128 + lane * 4) = r;    // global_store_b128
    }
}

extern "C" void kernel_launch(void* const* d_in, const int* in_sizes, int n_in,
                              void* d_out, int out_size, void* d_ws, size_t ws_size,
                              hipStream_t stream) {
    const float* x = (const float*)d_in[0];
    float* out = (float*)d_out;
    const int n    = in_sizes[0];
    const int rows = n / D2;                    // 16*2048 = 32768
    const int blocks = (rows + WAVES_PER_BLK - 1) / WAVES_PER_BLK;
    sparsemax1024_kernel<<<blocks, BLK_THREADS, 0, stream>>>(x, out, rows);
}